// MultiScaleConvCBAMNonLocalDecouple_38783554683413
// MI455X (gfx1250) — compile-verified
//
#include <hip/hip_runtime.h>

// MultiScaleConv + CBAM + Gaussian NonLocal, CDNA5 (gfx1250, wave32).
// fp32 WMMA (V_WMMA_F32_16X16X4_F32) for all GEMM-shaped work; flash-style
// streaming softmax for the non-local attention (no B*N*N buffer).

#define HW 1024   // 32x32 spatial
#define NB 16     // batch

typedef __attribute__((ext_vector_type(2))) float v2f;
typedef __attribute__((ext_vector_type(8))) float v8f;

__device__ __forceinline__ v8f wmma4(v2f a, v2f b, v8f c) {
  // D = A(16x4) * B(4x16) + C(16x16), fp32
  return __builtin_amdgcn_wmma_f32_16x16x4_f32(false, a, false, b, (short)0, c,
                                               false, false);
}

// ---------------------------------------------------------------------------
// Generic per-batch 1x1-conv GEMM: out[co,n] = sum_k W[co,k]*in[k,n] (+bias,+acc)
// One wave computes a 16(M) x 16*NT(N) tile; the A (weight) fragment is loaded
// once per K-chunk and reused across NT B fragments / WMMAs (register blocking
// raises wmma:vmem density and gives NT independent WMMAs to hide XDL latency).
// A layout: lanes 0-15 hold K=k0,k0+1 ; lanes 16-31 hold K=k0+2,k0+3 (row = lane&15)
// B layout: VGPR0 row k0(+2 for hi half), VGPR1 row k0+1(+3), col = lane&15
// C layout: acc[r] -> row r (lanes 0-15) / row r+8 (lanes 16-31), col = lane&15
// ---------------------------------------------------------------------------
template <int NT>
__global__ __launch_bounds__(32) void k_gemm1x1(
    const float* __restrict__ in, long inBS, const float* __restrict__ W,
    int Wld, const float* __restrict__ bias, float* __restrict__ out,
    long outBS, int K, int accumulate) {
  const int m0 = blockIdx.x << 4;
  const int n0 = blockIdx.y * (16 * NT);
  const int b = blockIdx.z;
  const float* ib = in + (long)b * inBS;
  float* ob = out + (long)b * outBS;
  const int lane = threadIdx.x;
  const int half = lane >> 4, l16 = lane & 15;
  const v8f vzero = {0.f, 0.f, 0.f, 0.f, 0.f, 0.f, 0.f, 0.f};
  v8f acc[NT];
#pragma unroll
  for (int nt = 0; nt < NT; ++nt) acc[nt] = vzero;
  const int arow = m0 + l16;
  for (int k0 = 0; k0 < K; k0 += 4) {
    const int ka = k0 + 2 * half;
    v2f a;
    a.x = W[arow * Wld + ka];
    a.y = W[arow * Wld + ka + 1];
    const float* r0 = ib + (long)ka * HW + n0 + l16;
    const float* r1 = ib + (long)(ka + 1) * HW + n0 + l16;
#pragma unroll
    for (int nt = 0; nt < NT; ++nt) {
      v2f bv;
      bv.x = r0[nt * 16];
      bv.y = r1[nt * 16];
      acc[nt] = wmma4(a, bv, acc[nt]);
    }
  }
#pragma unroll
  for (int nt = 0; nt < NT; ++nt)
#pragma unroll
    for (int r = 0; r < 8; ++r) {
      const int row = m0 + r + 8 * half;
      const long idx = (long)row * HW + n0 + nt * 16 + l16;
      float v = acc[nt][r];
      if (bias) v += bias[row];
      if (accumulate) v += ob[idx];
      ob[idx] = v;
    }
}

// ---------------------------------------------------------------------------
// Fused Gaussian non-local attention (flash-style, no NxN buffer).
// One wave handles a 16-query tile for one batch. CT = C/16 output tiles.
// out[i,c] = sum_j exp(exp(-max(|q_i|^2+|k_j|^2-2 q.k,0)/(2s^2))) * v[c,j] / rowsum
// Written to a flat (N,C) region (matches the torch view(B,C,H,W) reinterpret).
// ---------------------------------------------------------------------------
template <int CT>
__global__ __launch_bounds__(32) void k_nl_attn(
    const float* __restrict__ qp, const float* __restrict__ kp,
    const float* __restrict__ vp, const float* __restrict__ qq,
    const float* __restrict__ kk, const float* __restrict__ sigma,
    float* __restrict__ outBase, long outBS, int Ck) {
  __shared__ float qs[16 * 64];
  __shared__ float wt[16 * 17];
  const int b = blockIdx.y;
  const int i0 = blockIdx.x << 4;
  const int lane = threadIdx.x, half = lane >> 4, l16 = lane & 15;
  const int C = CT * 16;
  const float* qb = qp + (long)b * Ck * HW;
  const float* kb = kp + (long)b * Ck * HW;
  const float* vb = vp + (long)b * C * HW;
  const float* qqb = qq + b * HW;
  const float* kkb = kk + b * HW;
  const float s0 = sigma[0];
  const float inv2s2 = 1.0f / (2.0f * s0 * s0);

  for (int t = lane; t < 16 * Ck; t += 32) {   // 16*Ck is a multiple of 32
    const int ii = t / Ck, k = t - ii * Ck;
    qs[ii * 64 + k] = qb[(long)k * HW + i0 + ii];
  }
  __syncthreads();

  float qqr[8];
#pragma unroll
  for (int r = 0; r < 8; ++r) qqr[r] = qqb[i0 + r + 8 * half];
  const v8f vzero = {0.f, 0.f, 0.f, 0.f, 0.f, 0.f, 0.f, 0.f};
  v8f acc[CT];
#pragma unroll
  for (int t = 0; t < CT; ++t) acc[t] = vzero;
  float rs[8];
#pragma unroll
  for (int r = 0; r < 8; ++r) rs[r] = 0.f;

  for (int jt = 0; jt < 64; ++jt) {
    const int j0 = jt << 4;
    v8f s = vzero;
    for (int k0 = 0; k0 < Ck; k0 += 4) {   // S = Q * K^T  (16x16)
      const int ka = k0 + 2 * half;
      v2f a, bv;
      a.x = qs[l16 * 64 + ka];
      a.y = qs[l16 * 64 + ka + 1];
      bv.x = kb[(long)ka * HW + j0 + l16];
      bv.y = kb[(long)(ka + 1) * HW + j0 + l16];
      s = wmma4(a, bv, s);
    }
    const float kkc = kkb[j0 + l16];
    __syncthreads();
#pragma unroll
    for (int r = 0; r < 8; ++r) {
      float dist = qqr[r] + kkc - 2.0f * s[r];
      dist = dist < 0.f ? 0.f : dist;
      const float sim = __expf(-dist * inv2s2);
      const float w = __expf(sim);           // softmax numerator, bounded by e
      rs[r] += w;
      wt[(r + 8 * half) * 17 + l16] = w;     // stage C-layout -> LDS
    }
    __syncthreads();
#pragma unroll
    for (int kt = 0; kt < 4; ++kt) {         // out += W_tile * V_tile^T
      const int ka = (kt << 2) + 2 * half;
      v2f a;
      a.x = wt[l16 * 17 + ka];
      a.y = wt[l16 * 17 + ka + 1];
#pragma unroll
      for (int nt = 0; nt < CT; ++nt) {
        v2f bv;
        bv.x = vb[(long)(nt * 16 + l16) * HW + j0 + ka];
        bv.y = vb[(long)(nt * 16 + l16) * HW + j0 + ka + 1];
        acc[nt] = wmma4(a, bv, acc[nt]);
      }
    }
  }
  // reduce row sums across the 16 lanes of each half (butterfly keeps halves)
#pragma unroll
  for (int m = 1; m < 16; m <<= 1)
#pragma unroll
    for (int r = 0; r < 8; ++r) rs[r] += __shfl_xor(rs[r], m, 32);

  float* ob = outBase + (long)b * outBS;
#pragma unroll
  for (int nt = 0; nt < CT; ++nt)
#pragma unroll
    for (int r = 0; r < 8; ++r) {
      const int i = i0 + r + 8 * half;
      const int c = nt * 16 + l16;
      ob[(long)i * C + c] = acc[nt][r] / rs[r];
    }
}

// ---------------------------------------------------------------------------
// Elementwise / small helper kernels
// ---------------------------------------------------------------------------
__global__ void k_copych(float* __restrict__ dst, long dstBS,
                         const float* __restrict__ src, long srcBS, int n) {
  const int idx = blockIdx.x * blockDim.x + threadIdx.x;
  const int b = idx / n, t = idx - b * n;
  dst[(long)b * dstBS + t] = src[(long)b * srcBS + t];
}

__global__ void k_resadd(float* __restrict__ dst, long dstBS,
                         const float* __restrict__ src, long srcBS, int n) {
  const int idx = blockIdx.x * blockDim.x + threadIdx.x;
  const int b = idx / n, t = idx - b * n;
  dst[(long)b * dstBS + t] += src[(long)b * srcBS + t];
}

__global__ void k_conv3x3(const float* __restrict__ in, long inBS,
                          const float* __restrict__ w,
                          const float* __restrict__ bias,
                          float* __restrict__ out, long outBS) {
  const int idx = blockIdx.x * blockDim.x + threadIdx.x;
  const int p = idx & 1023;
  const int co = (idx >> 10) & 31;
  const int b = idx >> 15;
  const int h = p >> 5, x = p & 31;
  const float* ib = in + (long)b * inBS;
  float s = bias[co];
  for (int ci = 0; ci < 32; ++ci) {
    const float* ic = ib + (long)ci * HW;
    const float* wc = w + (co * 32 + ci) * 9;
#pragma unroll
    for (int kh = 0; kh < 3; ++kh) {
      const int hh = h + kh - 1;
      if (hh < 0 || hh > 31) continue;
#pragma unroll
      for (int kw = 0; kw < 3; ++kw) {
        const int ww = x + kw - 1;
        if (ww < 0 || ww > 31) continue;
        s += wc[kh * 3 + kw] * ic[hh * 32 + ww];
      }
    }
  }
  out[(long)b * outBS + (long)co * HW + p] = s;
}

__global__ __launch_bounds__(256) void k_chanstats(
    const float* __restrict__ in, long inBS, int C, float* __restrict__ avg,
    float* __restrict__ mxo) {
  const int bc = blockIdx.x;
  const int b = bc / C, c = bc - b * C;
  const float* p = in + (long)b * inBS + (long)c * HW;
  float s = 0.f, m = -3.4e38f;
  for (int t = threadIdx.x; t < HW; t += 256) {
    const float v = p[t];
    s += v;
    m = fmaxf(m, v);
  }
  __shared__ float ss[256], sm[256];
  ss[threadIdx.x] = s;
  sm[threadIdx.x] = m;
  __syncthreads();
  for (int o = 128; o > 0; o >>= 1) {
    if ((int)threadIdx.x < o) {
      ss[threadIdx.x] += ss[threadIdx.x + o];
      sm[threadIdx.x] = fmaxf(sm[threadIdx.x], sm[threadIdx.x + o]);
    }
    __syncthreads();
  }
  if (threadIdx.x == 0) {
    avg[bc] = ss[0] * (1.0f / HW);
    mxo[bc] = sm[0];
  }
}

__global__ __launch_bounds__(128) void k_cbam_mlp(
    const float* __restrict__ avg, const float* __restrict__ mx,
    const float* __restrict__ fc1, const float* __restrict__ fc2, int C, int r,
    float* __restrict__ ca) {
  const int b = blockIdx.x;
  __shared__ float ha[8], hm[8];
  const float* av = avg + b * C;
  const float* mv = mx + b * C;
  if ((int)threadIdx.x < r) {
    const int j = threadIdx.x;
    float sa = 0.f, sm = 0.f;
    for (int c = 0; c < C; ++c) {
      sa += fc1[j * C + c] * av[c];
      sm += fc1[j * C + c] * mv[c];
    }
    ha[j] = fmaxf(sa, 0.f);
    hm[j] = fmaxf(sm, 0.f);
  }
  __syncthreads();
  for (int c = threadIdx.x; c < C; c += 128) {
    float oa = 0.f, om = 0.f;
    for (int j = 0; j < r; ++j) {
      oa += fc2[c * r + j] * ha[j];
      om += fc2[c * r + j] * hm[j];
    }
    ca[b * C + c] = 1.f / (1.f + __expf(-(oa + om)));
  }
}

__global__ void k_cbam_apply1(const float* __restrict__ in, long inBS,
                              const float* __restrict__ ca, int C,
                              float* __restrict__ x1, long x1BS,
                              float* __restrict__ smap) {
  const int idx = blockIdx.x * blockDim.x + threadIdx.x;
  const int b = idx >> 10, p = idx & 1023;
  const float* cav = ca + b * C;
  float mean = 0.f, mx = -3.4e38f;
  for (int c = 0; c < C; ++c) {
    const float v = in[(long)b * inBS + (long)c * HW + p] * cav[c];
    x1[(long)b * x1BS + (long)c * HW + p] = v;
    mean += v;
    mx = fmaxf(mx, v);
  }
  smap[b * 2048 + p] = mean / (float)C;
  smap[b * 2048 + 1024 + p] = mx;
}

__global__ void k_cbam_apply2(const float* __restrict__ smap,
                              const float* __restrict__ w,
                              float* __restrict__ x1, long x1BS, int C) {
  const int idx = blockIdx.x * blockDim.x + threadIdx.x;
  const int b = idx >> 10, p = idx & 1023;
  const int h = p >> 5, x = p & 31;
  float s = 0.f;
  for (int ch = 0; ch < 2; ++ch) {
    const float* sp = smap + b * 2048 + ch * 1024;
    const float* wc = w + ch * 49;
    for (int kh = 0; kh < 7; ++kh) {
      const int hh = h + kh - 3;
      if (hh < 0 || hh > 31) continue;
      for (int kw = 0; kw < 7; ++kw) {
        const int ww = x + kw - 3;
        if (ww < 0 || ww > 31) continue;
        s += wc[kh * 7 + kw] * sp[hh * 32 + ww];
      }
    }
  }
  const float sa = 1.f / (1.f + __expf(-s));
  for (int c = 0; c < C; ++c) x1[(long)b * x1BS + (long)c * HW + p] *= sa;
}

__global__ void k_sqnorm(const float* __restrict__ p, float* __restrict__ o,
                         int Ck) {
  const int idx = blockIdx.x * blockDim.x + threadIdx.x;
  const int b = idx >> 10, i = idx & 1023;
  const float* pb = p + (long)b * Ck * HW;
  float s = 0.f;
  for (int k = 0; k < Ck; ++k) {
    const float v = pb[(long)k * HW + i];
    s += v * v;
  }
  o[idx] = s;
}

__global__ void k_dw(const float* __restrict__ in, const float* __restrict__ w,
                     const float* __restrict__ bias, float* __restrict__ out,
                     int ks) {
  const int idx = blockIdx.x * blockDim.x + threadIdx.x;
  const int p = idx & 1023;
  const int bc = idx >> 10;
  const int c = bc % 448;
  const int b = bc / 448;
  const int h = p >> 5, x = p & 31;
  const int hk = ks >> 1;
  const float* ic = in + ((long)b * 448 + c) * HW;
  const float* wc = w + c * ks * ks;
  float s = bias[c];
  for (int kh = 0; kh < ks; ++kh) {
    const int hh = h + kh - hk;
    if (hh < 0 || hh > 31) continue;
    for (int kw = 0; kw < ks; ++kw) {
      const int ww = x + kw - hk;
      if (ww < 0 || ww > 31) continue;
      s += wc[kh * ks + kw] * ic[hh * 32 + ww];
    }
  }
  out[((long)b * 448 + c) * HW + p] = s;
}

__global__ __launch_bounds__(256) void k_bnstats(const float* __restrict__ o,
                                                 float* __restrict__ mean,
                                                 float* __restrict__ var) {
  const int c = blockIdx.x;
  float s = 0.f, s2 = 0.f;
  for (int t = threadIdx.x; t < NB * HW; t += 256) {
    const int b = t >> 10, p = t & 1023;
    const float v = o[((long)b * 128 + c) * HW + p];
    s += v;
    s2 += v * v;
  }
  __shared__ float ss[256], s2s[256];
  ss[threadIdx.x] = s;
  s2s[threadIdx.x] = s2;
  __syncthreads();
  for (int off = 128; off > 0; off >>= 1) {
    if ((int)threadIdx.x < off) {
      ss[threadIdx.x] += ss[threadIdx.x + off];
      s2s[threadIdx.x] += s2s[threadIdx.x + off];
    }
    __syncthreads();
  }
  if (threadIdx.x == 0) {
    const float m = ss[0] * (1.0f / (NB * HW));
    mean[c] = m;
    var[c] = s2s[0] * (1.0f / (NB * HW)) - m * m;  // biased var
  }
}

__global__ void k_bnapply(const float* __restrict__ o,
                          const float* __restrict__ mean,
                          const float* __restrict__ var,
                          const float* __restrict__ g,
                          const float* __restrict__ be,
                          float* __restrict__ dst) {
  const int idx = blockIdx.x * blockDim.x + threadIdx.x;
  const int c = (idx >> 10) & 127;
  const float v = (o[idx] - mean[c]) * rsqrtf(var[c] + 1e-5f) * g[c] + be[c];
  dst[idx] = v > 0.f ? v : 0.f;
}

// ---------------------------------------------------------------------------
// Host orchestration
// ---------------------------------------------------------------------------
extern "C" void kernel_launch(void* const* d_in, const int* in_sizes, int n_in,
                              void* d_out, int out_size, void* d_ws,
                              size_t ws_size, hipStream_t stream) {
  (void)in_sizes; (void)out_size; (void)ws_size;
  if (n_in < 61) return;
  auto F = [&](int i) { return (const float*)d_in[i]; };
  const float* X = F(0);
  const float* SUBW = F(1);
  const float* SUBB = F(2);

  float* ws = (float*)d_ws;
  float* qcat = ws;                 // 16*448*1024
  float* qnext = ws + 7340032;      // 16*128*1024
  float* inp = ws + 9437184;        // 16*128*1024 (max Ci)
  float* acat = ws + 11534336;      // 16*256*1024 (max 2Ci)
  float* qp = ws + 15728640;        // 16*64*1024
  float* kp = ws + 16777216;        // 16*64*1024
  float* vp = ws + 17825792;        // 16*128*1024
  float* stat = ws + 19922944;
  float* avg = stat;                // 2048
  float* mxb = stat + 2048;         // 2048
  float* ca = stat + 4096;          // 2048
  float* qq = stat + 6144;          // 16384
  float* kk = stat + 22528;        // 16384
  float* smap = stat + 38912;       // 32768
  float* bnm = stat + 71680;        // 128
  float* bnv = stat + 71808;        // 128
  float* dtmp = inp;  // msc phase reuses inp+acat+qp region (16*448*1024)
  float* obuf = vp;   // msc phase reuses vp region (16*128*1024)

  constexpr int NT = 4;             // 16x64 tile per wave
  const int GY = HW / (16 * NT);    // 16 N-blocks

  for (int e = 0; e < 2; ++e) {
    const float* src = (e == 0) ? X : qnext;
    k_copych<<<NB * 128 * HW / 256, 256, 0, stream>>>(qcat, 448L * HW, src,
                                                      128L * HW, 128 * HW);
    int suboff = 128;
    for (int i = 0; i < 4; ++i) {
      const int Ci = 32 * (i + 1), Ck = Ci / 2, rr = Ci / 16;
      const float* fc1 = F(3 + 3 * i);
      const float* fc2 = F(4 + 3 * i);
      const float* saw = F(5 + 3 * i);
      const float* qw = F(15 + 7 * i);
      const float* qb2 = F(16 + 7 * i);
      const float* kw = F(17 + 7 * i);
      const float* kb2 = F(18 + 7 * i);
      const float* vw = F(19 + 7 * i);
      const float* vb2 = F(20 + 7 * i);
      const float* sg = F(21 + 7 * i);
      const float* tw = F(43 + 2 * i);
      const float* tb = F(44 + 2 * i);

      // inp = cat([subs[i-1], sc])
      if (i > 0)
        k_copych<<<NB * 32 * i * HW / 256, 256, 0, stream>>>(
            inp, (long)Ci * HW, qcat + (long)(suboff - 32 * i) * HW, 448L * HW,
            32 * i * HW);
      k_conv3x3<<<NB * 32 * HW / 256, 256, 0, stream>>>(
          qcat + (long)(32 * i) * HW, 448L * HW, SUBW, SUBB,
          inp + (long)(32 * i) * HW, (long)Ci * HW);

      // CBAM -> acat[0:Ci]
      k_chanstats<<<NB * Ci, 256, 0, stream>>>(inp, (long)Ci * HW, Ci, avg, mxb);
      k_cbam_mlp<<<NB, 128, 0, stream>>>(avg, mxb, fc1, fc2, Ci, rr, ca);
      k_cbam_apply1<<<NB * HW / 256, 256, 0, stream>>>(
          inp, (long)Ci * HW, ca, Ci, acat, 2L * Ci * HW, smap);
      k_cbam_apply2<<<NB * HW / 256, 256, 0, stream>>>(smap, saw, acat,
                                                       2L * Ci * HW, Ci);

      // NonLocal -> acat[Ci:2Ci] (flat (N,C) region == contiguous channels)
      dim3 gq(Ck / 16, GY, NB), gv(Ci / 16, GY, NB);
      k_gemm1x1<NT><<<gq, 32, 0, stream>>>(inp, (long)Ci * HW, qw, Ci, qb2, qp,
                                           (long)Ck * HW, Ci, 0);
      k_gemm1x1<NT><<<gq, 32, 0, stream>>>(inp, (long)Ci * HW, kw, Ci, kb2, kp,
                                           (long)Ck * HW, Ci, 0);
      k_gemm1x1<NT><<<gv, 32, 0, stream>>>(inp, (long)Ci * HW, vw, Ci, vb2, vp,
                                           (long)Ci * HW, Ci, 0);
      k_sqnorm<<<NB * HW / 256, 256, 0, stream>>>(qp, qq, Ck);
      k_sqnorm<<<NB * HW / 256, 256, 0, stream>>>(kp, kk, Ck);
      dim3 ga(64, NB);
      switch (i) {
        case 0: k_nl_attn<2><<<ga, 32, 0, stream>>>(qp, kp, vp, qq, kk, sg, acat + (long)Ci * HW, 2L * Ci * HW, Ck); break;
        case 1: k_nl_attn<4><<<ga, 32, 0, stream>>>(qp, kp, vp, qq, kk, sg, acat + (long)Ci * HW, 2L * Ci * HW, Ck); break;
        case 2: k_nl_attn<6><<<ga, 32, 0, stream>>>(qp, kp, vp, qq, kk, sg, acat + (long)Ci * HW, 2L * Ci * HW, Ck); break;
        default: k_nl_attn<8><<<ga, 32, 0, stream>>>(qp, kp, vp, qq, kk, sg, acat + (long)Ci * HW, 2L * Ci * HW, Ck); break;
      }
      k_resadd<<<NB * Ci * HW / 256, 256, 0, stream>>>(
          acat + (long)Ci * HW, 2L * Ci * HW, inp, (long)Ci * HW, Ci * HW);

      // tail 1x1 conv -> subs[i] region of qcat
      k_gemm1x1<NT><<<gv, 32, 0, stream>>>(acat, 2L * Ci * HW, tw, 2 * Ci, tb,
                                           qcat + (long)suboff * HW, 448L * HW,
                                           2 * Ci, 0);
      suboff += Ci;
    }

    // MSC: three depthwise scales, accumulated pointwise GEMM, BN+ReLU
    const float* dw1w = F(51); const float* dw1b = F(52);
    const float* dw3w = F(53); const float* dw3b = F(54);
    const float* dw5w = F(55); const float* dw5b = F(56);
    const float* pww = F(57);  const float* pwb = F(58);
    const float* bng = F(59);  const float* bnbt = F(60);
    float* dst = (e == 1) ? (float*)d_out : qnext;
    dim3 gp(8, GY, NB);
    k_dw<<<NB * 448 * HW / 256, 256, 0, stream>>>(qcat, dw1w, dw1b, dtmp, 1);
    k_gemm1x1<NT><<<gp, 32, 0, stream>>>(dtmp, 448L * HW, pww, 1344, pwb, obuf,
                                         128L * HW, 448, 0);
    k_dw<<<NB * 448 * HW / 256, 256, 0, stream>>>(qcat, dw3w, dw3b, dtmp, 3);
    k_gemm1x1<NT><<<gp, 32, 0, stream>>>(dtmp, 448L * HW, pww + 448, 1344,
                                         nullptr, obuf, 128L * HW, 448, 1);
    k_dw<<<NB * 448 * HW / 256, 256, 0, stream>>>(qcat, dw5w, dw5b, dtmp, 5);
    k_gemm1x1<NT><<<gp, 32, 0, stream>>>(dtmp, 448L * HW, pww + 896, 1344,
                                         nullptr, obuf, 128L * HW, 448, 1);
    k_bnstats<<<128, 256, 0, stream>>>(obuf, bnm, bnv);
    k_bnapply<<<NB * 128 * HW / 256, 256, 0, stream>>>(obuf, bnm, bnv, bng,
                                                       bnbt, dst);
  }
}